// GroupedCBOW_15822659519120
// MI455X (gfx1250) — compile-verified
//
#include <hip/hip_runtime.h>

typedef __attribute__((ext_vector_type(2))) float v2f;
typedef __attribute__((ext_vector_type(8))) float v8f;

#define BQ   16     // batch rows per block
#define LDA  304    // LDS row stride (floats), even -> 8B-aligned float2 reads
#define LL   10     // window length
#define DD   100    // embedding dim per attribute
#define KDIM 300    // concat dim

__global__ __launch_bounds__(256) void cbow_fused(
    const int* __restrict__ x, const int* __restrict__ lens,
    const float* __restrict__ Epoi, const float* __restrict__ Ehour, const float* __restrict__ Ewk,
    const float* __restrict__ Wpoi, const float* __restrict__ Bpoi,
    const float* __restrict__ Whour, const float* __restrict__ Bhour,
    const float* __restrict__ Wwk,  const float* __restrict__ Bwk,
    float* __restrict__ Opoi, float* __restrict__ Ohour, float* __restrict__ Owk)
{
    __shared__ float sA[BQ * LDA];     // pooled tile, 16 x 300 (padded)
    __shared__ int   sIdx[BQ * 32];    // 30 token indices per row (padded to 32)
    __shared__ int   sEff[BQ];
    __shared__ float sInv[BQ];

    const int tid    = threadIdx.x;
    const int block0 = blockIdx.x * BQ;

    // ---- stage token indices + effective lengths ----
    const int* xg = x + (size_t)block0 * (LL * 3);
    for (int i = tid; i < BQ * (LL * 3); i += 256) {
        sIdx[(i / 30) * 32 + (i % 30)] = xg[i];
    }
    if (tid < BQ) {
        int e = lens[block0 + tid];
        e = e < LL ? e : LL;
        sEff[tid] = e;
        sInv[tid] = 1.0f / (float)e;
    }
    __syncthreads();

    // ---- phase 1: ragged mean pooling into LDS (16 threads per row) ----
    {
        const int   r    = tid >> 4;
        const int   c    = tid & 15;
        const int   eff  = sEff[r];
        const float inv  = sInv[r];
        const int*  idxr = &sIdx[r * 32];
        for (int d = c; d < KDIM; d += 16) {
            const float* tab; int attr, dd;
            if (d < DD)            { tab = Epoi;  attr = 0; dd = d;          }
            else if (d < 2 * DD)   { tab = Ehour; attr = 1; dd = d - DD;     }
            else                   { tab = Ewk;   attr = 2; dd = d - 2 * DD; }
            float s = 0.f;
            for (int l = 0; l < eff; ++l) {
                s += tab[(size_t)idxr[l * 3 + attr] * DD + dd];
            }
            sA[r * LDA + d] = s * inv;
        }
    }
    __syncthreads();

    // ---- phase 2: fp32 WMMA GEMM, 16x16 output tiles over N = 1000|24|7 ----
    const int wave = tid >> 5;
    const int lane = tid & 31;
    const int m    = lane & 15;        // A row / B col / D col within tile
    const int half = lane >> 4;
    const int kb   = half << 1;        // K sub-offset for this half-wave

    // tiles: 0..62 -> poi (63 tiles cover 1008 >= 1000), 63..64 -> hour, 65 -> weekday
    for (int tile = wave; tile < 66; tile += 8) {
        int Nseg, colbase;
        const float *Wseg, *Bseg; float *Oseg;
        if (tile < 63)      { Nseg = 1000; colbase = tile * 16;        Wseg = Wpoi;  Bseg = Bpoi;  Oseg = Opoi;  }
        else if (tile < 65) { Nseg = 24;   colbase = (tile - 63) * 16; Wseg = Whour; Bseg = Bhour; Oseg = Ohour; }
        else                { Nseg = 7;    colbase = 0;                Wseg = Wwk;   Bseg = Bwk;   Oseg = Owk;   }

        const int    n    = colbase + m;
        const int    ncl  = n < Nseg ? n : (Nseg - 1);      // clamp OOB loads
        const float* Wrow = Wseg + (size_t)ncl * KDIM;      // B[k][n] = W[n][k]
        const float* Arow = &sA[m * LDA];

        v8f acc0 = {};
        v8f acc1 = {};
        int k = 0;
        #pragma unroll 4
        for (; k + 8 <= KDIM; k += 8) {
            v2f a0 = *(const v2f*)(Arow + k + kb);
            v2f b0 = *(const v2f*)(Wrow + k + kb);
            acc0 = __builtin_amdgcn_wmma_f32_16x16x4_f32(
                false, a0, false, b0, (short)0, acc0, false, false);
            v2f a1 = *(const v2f*)(Arow + k + 4 + kb);
            v2f b1 = *(const v2f*)(Wrow + k + 4 + kb);
            acc1 = __builtin_amdgcn_wmma_f32_16x16x4_f32(
                false, a1, false, b1, (short)0, acc1, false, false);
        }
        { // remainder: 300 % 8 == 4
            v2f a0 = *(const v2f*)(Arow + k + kb);
            v2f b0 = *(const v2f*)(Wrow + k + kb);
            acc0 = __builtin_amdgcn_wmma_f32_16x16x4_f32(
                false, a0, false, b0, (short)0, acc0, false, false);
        }
        v8f acc = acc0 + acc1;

        if (n < Nseg) {
            const float bias  = Bseg[n];
            const int   rbase = block0 + (half << 3);   // D row = j + 8*half
            #pragma unroll
            for (int j = 0; j < 8; ++j) {
                Oseg[(size_t)(rbase + j) * Nseg + n] = acc[j] + bias;
            }
        }
    }
}

extern "C" void kernel_launch(void* const* d_in, const int* in_sizes, int n_in,
                              void* d_out, int out_size, void* d_ws, size_t ws_size,
                              hipStream_t stream) {
    const int*   x     = (const int*)  d_in[0];
    const int*   lens  = (const int*)  d_in[1];
    const float* Epoi  = (const float*)d_in[2];
    const float* Ehour = (const float*)d_in[3];
    const float* Ewk   = (const float*)d_in[4];
    const float* Wpoi  = (const float*)d_in[5];
    const float* Bpoi  = (const float*)d_in[6];
    const float* Whour = (const float*)d_in[7];
    const float* Bhour = (const float*)d_in[8];
    const float* Wwk   = (const float*)d_in[9];
    const float* Bwk   = (const float*)d_in[10];

    const int B = in_sizes[1];               // lens has B elements
    float* out   = (float*)d_out;            // concat(poi[B,1000], hour[B,24], wk[B,7])
    float* Opoi  = out;
    float* Ohour = out + (size_t)B * 1000;
    float* Owk   = Ohour + (size_t)B * 24;

    dim3 grid(B / BQ), block(256);
    hipLaunchKernelGGL(cbow_fused, grid, block, 0, stream,
                       x, lens, Epoi, Ehour, Ewk,
                       Wpoi, Bpoi, Whour, Bhour, Wwk, Bwk,
                       Opoi, Ohour, Owk);
}